// Conv_54065048322391
// MI455X (gfx1250) — compile-verified
//
#include <hip/hip_runtime.h>

typedef __attribute__((ext_vector_type(2))) float v2f;
typedef __attribute__((ext_vector_type(8))) float v8f;

#define NN 100000
#define EE 1000000
#define DD 128

enum { EPI_NONE = 0, EPI_BIAS = 1, EPI_BIAS_PRELU = 2 };

// ---------------------------------------------------------------------------
// GEMM: Out[M x 128] = A[M x 128] @ W[128 x 128]  (+ optional bias / PReLU)
// 256 threads = 8 waves. W staged in 64KB LDS, pair-interleaved + XOR-swizzled
// so each lane's B fragment is ONE conflict-free ds_load_b64.
//   W[k][col] -> sW[(k>>1)*256 + ((col<<1) ^ (((k>>1)&1)<<5)) + (k&1)]
// Each wave computes a 16-row x 128-col slab: 8 v8f accumulators,
// 32 k-steps x 8 tiles of v_wmma_f32_16x16x4_f32.
// ---------------------------------------------------------------------------
__global__ __launch_bounds__(256)
void gemm128_wmma(const float* __restrict__ A, const float* __restrict__ W,
                  const float* __restrict__ bias, const float* __restrict__ alpha_p,
                  float* __restrict__ Out, int M, int mode)
{
    __shared__ float sW[DD * DD];   // 64 KB, pair-interleaved layout
    {
        const float4* Wv = (const float4*)W;
        for (int i = threadIdx.x; i < DD * DD / 4; i += 256) {
            const int k  = i >> 5;          // weight row
            const int c0 = (i & 31) << 2;   // first of 4 cols
            const float4 w = Wv[i];
            const int p    = k >> 1;
            const int base = p * 256 + (k & 1);
            const int sw   = (p & 1) << 5;
            sW[base + (((c0 + 0) << 1) ^ sw)] = w.x;
            sW[base + (((c0 + 1) << 1) ^ sw)] = w.y;
            sW[base + (((c0 + 2) << 1) ^ sw)] = w.z;
            sW[base + (((c0 + 3) << 1) ^ sw)] = w.w;
        }
    }
    __syncthreads();

    const int wave = threadIdx.x >> 5;
    const int lane = threadIdx.x & 31;
    const int half = lane >> 4;     // 0: lanes 0-15 (K=k,k+1), 1: lanes 16-31 (K=k+2,k+3)
    const int l16  = lane & 15;

    const int rowBase = blockIdx.x * 128 + wave * 16;
    int aRow = rowBase + l16;
    if (aRow >= M) aRow = M - 1;    // clamp loads; stores are predicated
    const float* Arow = A + (size_t)aRow * DD;

    // swizzle term is loop-invariant: pairIdx parity == half (k steps by 4)
    const int swz = half << 5;

    v8f acc[8] = {};

    for (int k = 0; k < DD; k += 4) {
        // A 16x4 fragment: one aligned 8-byte load per lane
        const v2f a = *(const v2f*)(Arow + k + 2 * half);
        const float* bbase = sW + ((k >> 1) + half) * 256;
        #pragma unroll
        for (int j = 0; j < 8; ++j) {
            const int col = j * 16 + l16;
            const v2f b = *(const v2f*)(bbase + (((col << 1) ^ swz)));
            acc[j] = __builtin_amdgcn_wmma_f32_16x16x4_f32(
                false, a, false, b, (short)0, acc[j], false, false);
        }
    }

    const float alpha = (mode == EPI_BIAS_PRELU) ? alpha_p[0] : 0.0f;
    #pragma unroll
    for (int j = 0; j < 8; ++j) {
        const int col = j * 16 + l16;
        const float bv = (mode != EPI_NONE) ? bias[col] : 0.0f;
        #pragma unroll
        for (int r = 0; r < 8; ++r) {
            const int row = rowBase + r + 8 * half;   // C layout: VGPR r -> M = r + 8*half
            if (row < M) {
                float v = acc[j][r] + bv;
                if (mode == EPI_BIAS_PRELU) v = (v > 0.0f) ? v : alpha * v;
                Out[(size_t)row * DD + col] = v;
            }
        }
    }
}

// ---------------------------------------------------------------------------
// Degree / normalization
// ---------------------------------------------------------------------------
__global__ void k_deg_init(float* __restrict__ deg) {
    int i = blockIdx.x * blockDim.x + threadIdx.x;
    if (i < NN) deg[i] = 1.0f;                       // self-loop
}

__global__ void k_deg_count(const int* __restrict__ dst, float* __restrict__ deg) {
    int e = blockIdx.x * blockDim.x + threadIdx.x;
    if (e < EE) atomicAdd(&deg[dst[e]], 1.0f);
}

__global__ void k_rsqrt(float* __restrict__ deg) {
    int i = blockIdx.x * blockDim.x + threadIdx.x;
    if (i < NN) deg[i] = rsqrtf(deg[i]);             // in place -> dinv
}

// out[i,:] = dinv[i]^2 * h[i,:]   (self-loop term; also initializes out)
__global__ void k_self(const float* __restrict__ h, const float* __restrict__ nrm,
                       float* __restrict__ out) {
    int t = blockIdx.x * blockDim.x + threadIdx.x;
    int i = t >> 5, f = (t & 31) << 2;
    if (i >= NN) return;
    float w = nrm[i] * nrm[i];
    float4 v = *(const float4*)(h + (size_t)i * DD + f);
    v.x *= w; v.y *= w; v.z *= w; v.w *= w;
    *(float4*)(out + (size_t)i * DD + f) = v;
}

// per-edge: out[dst,:] += dinv[src]*dinv[dst] * h[src,:]   (one wave per edge)
__global__ void k_scatter(const int* __restrict__ src, const int* __restrict__ dst,
                          const float* __restrict__ h, const float* __restrict__ nrm,
                          float* __restrict__ out) {
    long long t = (long long)blockIdx.x * blockDim.x + threadIdx.x;
    int e = (int)(t >> 5), f = ((int)t & 31) << 2;
    if (e >= EE) return;
    const int s = src[e], d = dst[e];
    const float w = nrm[s] * nrm[d];
    float4 v = *(const float4*)(h + (size_t)s * DD + f);
    float* o = out + (size_t)d * DD + f;
    atomicAdd(o + 0, w * v.x);
    atomicAdd(o + 1, w * v.y);
    atomicAdd(o + 2, w * v.z);
    atomicAdd(o + 3, w * v.w);
}

__global__ void k_bias_relu(float* __restrict__ io, const float* __restrict__ b) {
    int t = blockIdx.x * blockDim.x + threadIdx.x;
    int i = t >> 5, f = (t & 31) << 2;
    if (i >= NN) return;
    float4 v = *(const float4*)(io + (size_t)i * DD + f);
    v.x = fmaxf(v.x + b[f + 0], 0.0f);
    v.y = fmaxf(v.y + b[f + 1], 0.0f);
    v.z = fmaxf(v.z + b[f + 2], 0.0f);
    v.w = fmaxf(v.w + b[f + 3], 0.0f);
    *(float4*)(io + (size_t)i * DD + f) = v;
}

// ---------------------------------------------------------------------------
extern "C" void kernel_launch(void* const* d_in, const int* in_sizes, int n_in,
                              void* d_out, int out_size, void* d_ws, size_t ws_size,
                              hipStream_t stream) {
    const float* x   = (const float*)d_in[0];
    const int*   ei  = (const int*)d_in[1];
    const int*   src = ei;
    const int*   dst = ei + EE;
    const float* W1  = (const float*)d_in[2];
    const float* b1  = (const float*)d_in[3];
    const float* W2  = (const float*)d_in[4];
    const float* b2  = (const float*)d_in[5];
    const float* Wp1 = (const float*)d_in[6];
    const float* bp1 = (const float*)d_in[7];
    const float* pa  = (const float*)d_in[8];
    const float* Wp2 = (const float*)d_in[9];
    const float* bp2 = (const float*)d_in[10];

    float* zs  = (float*)d_out;               // z: [N,128]
    float* ps  = zs + (size_t)NN * DD;        // p: [N,128]
    float* nrm = (float*)d_ws;                // [N]
    float* bufA = nrm + NN;                   // [N,128]
    float* bufB = bufA + (size_t)NN * DD;     // [N,128]

    dim3 blk(256);
    const int gN  = (NN + 255) / 256;
    const int gE  = (EE + 255) / 256;
    const int gNv = (NN * 32 + 255) / 256;
    const int gEv = (int)(((long long)EE * 32 + 255) / 256);
    const int gG  = (NN + 127) / 128;

    // normalization coefficients
    k_deg_init <<<gN, blk, 0, stream>>>(nrm);
    k_deg_count<<<gE, blk, 0, stream>>>(dst, nrm);
    k_rsqrt    <<<gN, blk, 0, stream>>>(nrm);

    // layer 1: z1 = relu(Ahat @ (x @ W1) + b1)
    gemm128_wmma<<<gG, blk, 0, stream>>>(x, W1, nullptr, nullptr, bufA, NN, EPI_NONE);
    k_self   <<<gNv, blk, 0, stream>>>(bufA, nrm, bufB);
    k_scatter<<<gEv, blk, 0, stream>>>(src, dst, bufA, nrm, bufB);
    k_bias_relu<<<gNv, blk, 0, stream>>>(bufB, b1);

    // layer 2: z = relu(Ahat @ (z1 @ W2) + b2)   -> first output
    gemm128_wmma<<<gG, blk, 0, stream>>>(bufB, W2, nullptr, nullptr, bufA, NN, EPI_NONE);
    k_self   <<<gNv, blk, 0, stream>>>(bufA, nrm, zs);
    k_scatter<<<gEv, blk, 0, stream>>>(src, dst, bufA, nrm, zs);
    k_bias_relu<<<gNv, blk, 0, stream>>>(zs, b2);

    // projection head: p = prelu(z @ Wp1 + bp1) @ Wp2 + bp2  -> second output
    gemm128_wmma<<<gG, blk, 0, stream>>>(zs, Wp1, bp1, pa, bufA, NN, EPI_BIAS_PRELU);
    gemm128_wmma<<<gG, blk, 0, stream>>>(bufA, Wp2, bp2, nullptr, ps, NN, EPI_BIAS);
}